// TokenFromSubtoken_4827543240879
// MI455X (gfx1250) — compile-verified
//
#include <hip/hip_runtime.h>

// TokenFromSubtoken: pack rows of units[b,s,:] where mask[b,s]==1 left-aligned
// into out[b,t,:], zero-padded to T = out_size/(B*F).
//
// Phase 1: one 1024-thread block per batch row; int4 mask load + hierarchical
//          wave32 shfl scan -> src_idx[b][t] (source subtoken of t-th token)
//          and lens[b] in workspace. One memory round-trip per row.
// Phase 2: one 256-thread block per 4 output rows; gather-copy 4KB rows
//          through LDS with CDNA5 async global<->LDS ops (ASYNCcnt), four
//          loads in flight per wave before a single s_wait_asynccnt; pad rows
//          zero-filled with 128-bit stores.

#define SCAN_BLOCK 1024
#define PACK_BLOCK 256
#define ROWS_PER_BLOCK 4

__global__ void __launch_bounds__(SCAN_BLOCK)
tfs_build_index(const int* __restrict__ mask,
                int* __restrict__ src_idx,
                int* __restrict__ lens,
                int S, int T) {
  const int b    = blockIdx.x;
  const int tid  = (int)threadIdx.x;
  const int lane = tid & 31;     // wave32
  const int wave = tid >> 5;     // 0..31
  const int* __restrict__ mrow = mask + (size_t)b * S;
  int* __restrict__ irow = src_idx + (size_t)b * T;

  __shared__ int wave_tot[32];
  __shared__ int wave_off[32];
  __shared__ int tot_sh;

  int running = 0;
  const int CH = SCAN_BLOCK * 4;                 // elements per block pass
  for (int base0 = 0; base0 < S; base0 += CH) {
    const int base = base0 + tid * 4;
    int v0 = 0, v1 = 0, v2 = 0, v3 = 0;
    if (base + 4 <= S) {
      const int4 m = *reinterpret_cast<const int4*>(mrow + base);
      v0 = (m.x != 0); v1 = (m.y != 0); v2 = (m.z != 0); v3 = (m.w != 0);
    } else {
      if (base + 0 < S) v0 = (mrow[base + 0] != 0);
      if (base + 1 < S) v1 = (mrow[base + 1] != 0);
      if (base + 2 < S) v2 = (mrow[base + 2] != 0);
      if (base + 3 < S) v3 = (mrow[base + 3] != 0);
    }
    const int cnt = v0 + v1 + v2 + v3;

    // inclusive scan of per-thread counts within the wave
    int scan = cnt;
#pragma unroll
    for (int d = 1; d < 32; d <<= 1) {
      const int n = __shfl_up(scan, d);
      if (lane >= d) scan += n;
    }

    __syncthreads();                     // protect smem reuse across passes
    if (lane == 31) wave_tot[wave] = scan;
    __syncthreads();

    if (wave == 0) {                     // scan the 32 wave totals
      const int wtot = wave_tot[lane];
      int wscan = wtot;
#pragma unroll
      for (int d = 1; d < 32; d <<= 1) {
        const int n = __shfl_up(wscan, d);
        if (lane >= d) wscan += n;
      }
      wave_off[lane] = wscan - wtot;     // exclusive wave offset
      if (lane == 31) tot_sh = wscan;    // block total this pass
    }
    __syncthreads();

    int p = running + wave_off[wave] + (scan - cnt);  // exclusive thread prefix
    if (v0) irow[p++] = base + 0;
    if (v1) irow[p++] = base + 1;
    if (v2) irow[p++] = base + 2;
    if (v3) irow[p++] = base + 3;
    running += tot_sh;
  }
  if (tid == 0) lens[b] = running;
}

__global__ void __launch_bounds__(PACK_BLOCK)
tfs_pack(const float* __restrict__ units,
         const int* __restrict__ src_idx,
         const int* __restrict__ lens,
         float* __restrict__ out,
         int S, int T, int F) {
  const int t0  = (int)blockIdx.x * ROWS_PER_BLOCK;
  const int b   = (int)blockIdx.y;
  const int len = lens[b];
  const int nrows = (T - t0 < ROWS_PER_BLOCK) ? (T - t0) : ROWS_PER_BLOCK;

  // 4 disjoint 4KB LDS regions (one per token row); each wave owns disjoint
  // 16B/lane slots, so only the wave's own ASYNCcnt needs draining.
  __shared__ char smem[ROWS_PER_BLOCK * PACK_BLOCK * 16];
  const unsigned ldsBase =
      (unsigned)(size_t)(&smem[0]) + (unsigned)threadIdx.x * 16u;

  const float* srow[ROWS_PER_BLOCK];
  bool inT[ROWS_PER_BLOCK];
#pragma unroll
  for (int j = 0; j < ROWS_PER_BLOCK; ++j) {
    const int t = t0 + j;
    inT[j]  = (j < nrows);
    srow[j] = nullptr;
    if (inT[j] && t < len) {
      const int s = src_idx[(size_t)b * T + t];
      srow[j] = units + ((size_t)b * S + s) * (size_t)F;
    }
  }

  for (int i0 = 0; i0 < F; i0 += PACK_BLOCK * 4) {
    const int i = i0 + (int)threadIdx.x * 4;
    if (i0 > 0)  // drain previous chunk's stores before overwriting LDS
      asm volatile("s_wait_asynccnt 0" ::: "memory");

    // Issue up to ROWS_PER_BLOCK async loads back-to-back (memory -> LDS).
#pragma unroll
    for (int j = 0; j < ROWS_PER_BLOCK; ++j) {
      if (srow[j]) {
        const float* sp = srow[j] + i;
        const unsigned lo = ldsBase + (unsigned)(j * PACK_BLOCK * 16);
        asm volatile("global_load_async_to_lds_b128 %0, %1, off"
                     :: "v"(lo), "v"(sp) : "memory");
      }
    }
    asm volatile("s_wait_asynccnt 0" ::: "memory");

    // Issue the stores (LDS -> memory) / zero-fill pad rows.
#pragma unroll
    for (int j = 0; j < ROWS_PER_BLOCK; ++j) {
      if (!inT[j]) continue;
      float* dp = out + ((size_t)b * T + (size_t)(t0 + j)) * (size_t)F + i;
      if (srow[j]) {
        const unsigned lo = ldsBase + (unsigned)(j * PACK_BLOCK * 16);
        asm volatile("global_store_async_from_lds_b128 %0, %1, off"
                     :: "v"(dp), "v"(lo) : "memory");
      } else {
        *reinterpret_cast<float4*>(dp) = make_float4(0.f, 0.f, 0.f, 0.f);
      }
    }
  }
  // Final async stores are drained by the implicit wait-idle at s_endpgm.
}

extern "C" void kernel_launch(void* const* d_in, const int* in_sizes, int n_in,
                              void* d_out, int out_size, void* d_ws, size_t ws_size,
                              hipStream_t stream) {
  (void)n_in; (void)ws_size;
  const float* units = (const float*)d_in[0];   // [B, S, F] float32
  const int*   mask  = (const int*)d_in[1];     // [B, S]    int32

  const int B  = 16;                        // fixed by reference
  const int BS = in_sizes[1];               // B*S
  const int S  = BS / B;
  const int F  = in_sizes[0] / BS;          // 1024
  const int T  = out_size / (B * F);        // harness-provided pad length

  int* src_idx = (int*)d_ws;                // [B, T]
  int* lens    = src_idx + (size_t)B * T;   // [B]

  tfs_build_index<<<dim3(B), dim3(SCAN_BLOCK), 0, stream>>>(mask, src_idx,
                                                            lens, S, T);

  dim3 grid((unsigned)((T + ROWS_PER_BLOCK - 1) / ROWS_PER_BLOCK), (unsigned)B);
  tfs_pack<<<grid, dim3(PACK_BLOCK), 0, stream>>>(units, src_idx, lens,
                                                  (float*)d_out, S, T, F);
}